// Softmax_30648886624542
// MI455X (gfx1250) — compile-verified
//
#include <hip/hip_runtime.h>

// Softmax (unstabilized, matching reference): out = exp(x) / sum(exp(x))
// N = 33,554,432 fp32 (128 MB). Memory-bound.
//
// MI455X strategy:
//  - Pass 1 (reduce): RT loads -> input becomes resident in the 192MB L2.
//  - Pass 2 (normalize): re-read input (L2 hits), write output with
//    NON-TEMPORAL stores so the 128MB output stream does not evict the input.
//    HBM traffic ~256MB -> ~11us at 23.3 TB/s.
//  - wave32 reductions use V_WMMA_F32_16X16X4_F32 with B=ones:
//    D[m][n] = sum_k A[m][k]; A layout (16x4 f32) holds lane m at K=0 and
//    lane m+16 at K=2, so one WMMA produces s_m + s_{m+16} replicated across
//    columns. Summing the 8 C/D VGPRs then one shfl_xor(16) yields the full
//    32-lane sum. EXEC is all-ones at every WMMA call site.

typedef float v2f __attribute__((ext_vector_type(2)));
typedef float v4f __attribute__((ext_vector_type(4)));
typedef float v8f __attribute__((ext_vector_type(8)));

#define RED_BLOCKS 2048
#define BLOCK      256
#define NORM_BLOCKS 8192

__device__ __forceinline__ float wave_reduce_sum(float s) {
    // A (16x4 f32, 2 VGPRs/lane): lanes 0-15 -> {K0=s, K1=0}, lanes 16-31 -> {K2=s, K3=0}
    v2f a; a[0] = s;    a[1] = 0.0f;
    // B = all ones (layout-independent)
    v2f b; b[0] = 1.0f; b[1] = 1.0f;
    v8f c = {};
    // D[m][n] = s_m + s_{m+16}, replicated across all 16 columns.
    v8f d = __builtin_amdgcn_wmma_f32_16x16x4_f32(
        /*neg_a=*/false, a, /*neg_b=*/false, b,
        /*c_mod=*/(short)0, c, /*reuse_a=*/false, /*reuse_b=*/false);
    // Lanes 0-15 hold rows M=0..7, lanes 16-31 hold rows M=8..15.
    float t = ((d[0] + d[1]) + (d[2] + d[3])) + ((d[4] + d[5]) + (d[6] + d[7]));
    // Combine the two half-sums -> full 32-lane sum in every lane.
    t += __shfl_xor(t, 16, 32);
    return t;
}

__device__ __forceinline__ float block_reduce_sum(float v) {
    __shared__ float lds[BLOCK / 32];
    const int lane = threadIdx.x & 31;
    const int wave = threadIdx.x >> 5;
    float w = wave_reduce_sum(v);          // whole wave active: EXEC all ones
    if (lane == 0) lds[wave] = w;
    __syncthreads();
    float t = 0.0f;
    if (wave == 0) {                       // wave 0 fully active inside branch
        float x = (lane < (int)(blockDim.x >> 5)) ? lds[lane] : 0.0f;
        t = wave_reduce_sum(x);
    }
    return t;                              // valid in wave 0 (all its lanes)
}

// Pass 1: per-block partial sums of exp(x). Regular (RT) loads warm L2.
__global__ void softmax_reduce_kernel(const float* __restrict__ in,
                                      float* __restrict__ partials,
                                      int nvec, int n) {
    const v4f* __restrict__ in4 = (const v4f*)in;
    float s = 0.0f;
    const int stride = gridDim.x * blockDim.x;
    for (int i = blockIdx.x * blockDim.x + threadIdx.x; i < nvec; i += stride) {
        v4f x = in4[i];                    // global_load_b128
        s += (__expf(x[0]) + __expf(x[1])) + (__expf(x[2]) + __expf(x[3]));
    }
    // Scalar tail (N is a multiple of 4 here, but stay generic).
    if (blockIdx.x == 0 && threadIdx.x == 0) {
        for (int i = nvec * 4; i < n; ++i) s += __expf(in[i]);
    }
    float t = block_reduce_sum(s);
    if (threadIdx.x == 0) partials[blockIdx.x] = t;
}

// Pass 1b: fold RED_BLOCKS partials, publish 1/sum.
__global__ void softmax_finalize_kernel(const float* __restrict__ partials,
                                        float* __restrict__ invp) {
    float s = 0.0f;
    for (int i = threadIdx.x; i < RED_BLOCKS; i += BLOCK) s += partials[i];
    float t = block_reduce_sum(s);
    if (threadIdx.x == 0) invp[0] = 1.0f / t;
}

// Pass 2: out = exp(x) * inv. Input loads hit L2; NT stores keep the
// output stream from evicting the L2-resident input.
__global__ void softmax_normalize_kernel(const float* __restrict__ in,
                                         float* __restrict__ out,
                                         const float* __restrict__ invp,
                                         int nvec, int n) {
    const float inv = invp[0];             // uniform; cached
    const v4f* __restrict__ in4 = (const v4f*)in;
    v4f* __restrict__ out4 = (v4f*)out;
    const int stride = gridDim.x * blockDim.x;
    for (int i = blockIdx.x * blockDim.x + threadIdx.x; i < nvec; i += stride) {
        v4f x = in4[i];
        v4f r;
        r[0] = __expf(x[0]) * inv;
        r[1] = __expf(x[1]) * inv;
        r[2] = __expf(x[2]) * inv;
        r[3] = __expf(x[3]) * inv;
        __builtin_nontemporal_store(r, out4 + i);   // global_store_b128 th:NT
    }
    if (blockIdx.x == 0 && threadIdx.x == 0) {
        for (int i = nvec * 4; i < n; ++i) out[i] = __expf(in[i]) * inv;
    }
}

extern "C" void kernel_launch(void* const* d_in, const int* in_sizes, int n_in,
                              void* d_out, int out_size, void* d_ws, size_t ws_size,
                              hipStream_t stream) {
    const float* in  = (const float*)d_in[0];
    float*       out = (float*)d_out;
    const int n    = in_sizes[0];
    const int nvec = n >> 2;

    float* partials = (float*)d_ws;          // RED_BLOCKS floats
    float* invp     = partials + RED_BLOCKS; // 1 float
    (void)ws_size; (void)n_in; (void)out_size;

    softmax_reduce_kernel<<<RED_BLOCKS, BLOCK, 0, stream>>>(in, partials, nvec, n);
    softmax_finalize_kernel<<<1, BLOCK, 0, stream>>>(partials, invp);
    softmax_normalize_kernel<<<NORM_BLOCKS, BLOCK, 0, stream>>>(in, out, invp, nvec, n);
}